// SPCModel_46806553592374
// MI455X (gfx1250) — compile-verified
//
#include <hip/hip_runtime.h>
#include <stdint.h>

// SPC model forward: orthonormal FWHT over N = 512*512 per (batch,channel),
// then diagonal mask multiply.  Separable form on the 512x512 image:
//   Y = (1/512) * H512 * X * H512 ;  out = mask .* Y
// H512 = H16 (x) H32  ->  per 512-vector: FWHT-32 on contiguous 32-blocks
// (register butterflies, exact) + FWHT-16 across blocks as a 16x16 matmul
// with the +/-1 Hadamard matrix via 4 chained V_WMMA_F32_16X16X4_F32 (exact
// f32 arithmetic).  Memory-bound problem -> async global->LDS staging.

typedef float v2f __attribute__((ext_vector_type(2)));
typedef float v8f __attribute__((ext_vector_type(8)));

#define IM 512
#define NP (IM * IM)
#define SCALE (1.0f / 512.0f)   // 1/sqrt(2^18)

__device__ __forceinline__ float h16sign(int m, int k) {
  return (__popc(m & k) & 1) ? -1.0f : 1.0f;
}

template <int H>
__device__ __forceinline__ void bfly32(float* r) {
#pragma unroll
  for (int i = 0; i < 32; i += 2 * H) {
#pragma unroll
    for (int j = 0; j < H; ++j) {
      float a = r[i + j], b = r[i + j + H];
      r[i + j] = a + b;
      r[i + j + H] = a - b;
    }
  }
}

// Apply unnormalized FWHT-512 to 16 segments of 512 floats held in LDS.
// Segment s element e lives at tile[s*P + e].  128 threads (4 waves).
template <int P>
__device__ __forceinline__ void fwht512_tile(float* tile, int tid) {
  // ---- Stage A: FWHT-32 on each contiguous 32-block, in registers ----
#pragma unroll
  for (int bb = 0; bb < 2; ++bb) {
    const int blk = tid + bb * 128;           // 256 blocks = (segment s, u)
    const int s = blk >> 4, u = blk & 15;
    float* p = &tile[s * P + u * 32];
    float r[32];
#pragma unroll
    for (int i = 0; i < 8; ++i) {
      float4 v = *(float4*)&p[4 * i];
      r[4 * i + 0] = v.x; r[4 * i + 1] = v.y;
      r[4 * i + 2] = v.z; r[4 * i + 3] = v.w;
    }
    bfly32<1>(r); bfly32<2>(r); bfly32<4>(r); bfly32<8>(r); bfly32<16>(r);
#pragma unroll
    for (int i = 0; i < 8; ++i) {
      float4 v = make_float4(r[4 * i], r[4 * i + 1], r[4 * i + 2], r[4 * i + 3]);
      *(float4*)&p[4 * i] = v;
    }
  }
  __syncthreads();

  // ---- Stage B: FWHT-16 across the 16 sub-blocks (stride 32) ----
  // Out[u'][col] = sum_u H16[u'][u] * Mid[u][col]  ==  D = H16 x B,
  // as 4 chained v_wmma_f32_16x16x4_f32 (K = u, split in 4 chunks of 4).
  const int lane = tid & 31;
  const int wave = tid >> 5;
  const int n = lane & 15;            // M (for A) / N (for B,D)
  const int hi = lane >> 4;           // half-wave selects K pair / M+8

  // A = H16 chunk kc: VGPR0 holds K=4kc+2*hi, VGPR1 holds K=4kc+1+2*hi
  v2f A[4];
#pragma unroll
  for (int kc = 0; kc < 4; ++kc) {
    const int kb = 4 * kc + 2 * hi;
    A[kc].x = h16sign(n, kb);
    A[kc].y = h16sign(n, kb + 1);
  }

  // 32 groups: (segment s, column half v0) ; 8 per wave; each group is
  // a full 16x16x16 product chained from 4 K=4 WMMAs.
#pragma unroll
  for (int g = 0; g < 8; ++g) {
    const int grp = wave * 8 + g;
    const int s = grp >> 1;
    const int v0 = (grp & 1) * 16;
    float* seg = &tile[s * P];
    const int col = v0 + n;

    v8f acc = {};
#pragma unroll
    for (int kc = 0; kc < 4; ++kc) {
      v2f B;                                   // B[K=u][N=col]
      B.x = seg[(4 * kc + 2 * hi + 0) * 32 + col];
      B.y = seg[(4 * kc + 2 * hi + 1) * 32 + col];
      acc = __builtin_amdgcn_wmma_f32_16x16x4_f32(
          false, A[kc], false, B, (short)0, acc, false, false);
    }
    // D layout: vgpr m -> M = m + 8*hi, N = lane&15
#pragma unroll
    for (int m = 0; m < 8; ++m) {
      seg[(m + 8 * hi) * 32 + col] = acc[m];
    }
  }
  __syncthreads();
}

// ---- Pass 1: FWHT along the contiguous axis (each of 48*512 segments) ----
__global__ __launch_bounds__(128) void spc_fwht_rows(
    const float* __restrict__ x, float* __restrict__ out) {
  __shared__ __attribute__((aligned(16))) float tile[16 * 512];
  const int tid = threadIdx.x;
  const long long base = (long long)blockIdx.x * (16 * 512);
  const float* src = x + base;

  // async global -> LDS, 16B per lane per op
  const unsigned lds0 = (unsigned)(uintptr_t)&tile[0];
#pragma unroll 4
  for (int i = 0; i < 16; ++i) {
    const int f = tid + 128 * i;                 // float4 index
    const unsigned lds = lds0 + (unsigned)f * 16u;
    const unsigned gof = (unsigned)f * 16u;
    asm volatile("global_load_async_to_lds_b128 %0, %1, %2"
                 :: "v"(lds), "v"(gof), "s"(src) : "memory");
  }
  asm volatile("s_wait_asynccnt 0" ::: "memory");
  __syncthreads();

  fwht512_tile<512>(tile, tid);

  float4* __restrict__ dst = (float4*)(out + base);
  const float4* tv = (const float4*)tile;
#pragma unroll 4
  for (int i = 0; i < 16; ++i) {
    const int f = tid + 128 * i;
    dst[f] = tv[f];
  }
}

// ---- Pass 2: FWHT along the stride-512 axis, fused mask * 1/512, in-place --
__global__ __launch_bounds__(128) void spc_fwht_cols_mask(
    float* __restrict__ io, const float* __restrict__ mask) {
  constexpr int P = 516;                      // padded pitch (bank spread)
  __shared__ __attribute__((aligned(16))) float tile[16 * P];
  const int tid = threadIdx.x;
  const int img = blockIdx.x >> 5;            // 48 images
  const int c0 = (blockIdx.x & 31) * 16;      // 16-column strip
  float* base = io + (long long)img * NP + c0;

  // async transpose-load: element (r, c0+cc) -> tile[cc*P + r]
  const unsigned lds0 = (unsigned)(uintptr_t)&tile[0];
#pragma unroll 4
  for (int i = 0; i < 64; ++i) {
    const int e = tid + 128 * i;              // 8192 scalars
    const int r = e >> 4, cc = e & 15;
    const unsigned lds = lds0 + (unsigned)(cc * P + r) * 4u;
    const unsigned gof = (unsigned)(r * IM + cc) * 4u;
    asm volatile("global_load_async_to_lds_b32 %0, %1, %2"
                 :: "v"(lds), "v"(gof), "s"(base) : "memory");
  }
  asm volatile("s_wait_asynccnt 0" ::: "memory");
  __syncthreads();

  fwht512_tile<P>(tile, tid);

  // epilogue: out[r*512 + c0+cc] = tile[cc*P + r] * mask * 1/512
  const float* mbase = mask + c0;
#pragma unroll 4
  for (int i = 0; i < 16; ++i) {
    const int f = tid + 128 * i;              // float4 index over 512x16
    const int r = f >> 2, q = f & 3;
    const float4 m4 = *(const float4*)&mbase[r * IM + q * 4];
    float4 v;
    v.x = tile[(q * 4 + 0) * P + r] * m4.x * SCALE;
    v.y = tile[(q * 4 + 1) * P + r] * m4.y * SCALE;
    v.z = tile[(q * 4 + 2) * P + r] * m4.z * SCALE;
    v.w = tile[(q * 4 + 3) * P + r] * m4.w * SCALE;
    *(float4*)&base[r * IM + q * 4] = v;
  }
}

extern "C" void kernel_launch(void* const* d_in, const int* in_sizes, int n_in,
                              void* d_out, int out_size, void* d_ws, size_t ws_size,
                              hipStream_t stream) {
  (void)in_sizes; (void)n_in; (void)out_size; (void)d_ws; (void)ws_size;
  const float* x = (const float*)d_in[0];      // [16,3,512,512] f32
  const float* mask = (const float*)d_in[1];   // [262144] f32
  float* out = (float*)d_out;                  // [16,3,262144] f32

  // Pass 1: 48*512 contiguous 512-segments, 16 per block.
  spc_fwht_rows<<<dim3(1536), dim3(128), 0, stream>>>(x, out);
  // Pass 2: 48 images * 32 column-strips, in-place on d_out, fused mask.
  spc_fwht_cols_mask<<<dim3(1536), dim3(128), 0, stream>>>(out, mask);
}